// Decoder_23983097381305
// MI455X (gfx1250) — compile-verified
//
#include <hip/hip_runtime.h>
#include <math.h>

typedef __attribute__((ext_vector_type(2))) float v2f;
typedef __attribute__((ext_vector_type(8))) float v8f;

static constexpr int HD  = 2048;    // hidden dim
static constexpr int VOC = 50257;   // vocab
static constexpr int SEQ = 64;      // encoder positions

// workspace layout (float offsets)
static constexpr int WS_XCAT   = 0;       // 4096: [embedded | attn_applied]
static constexpr int WS_XVEC   = 4096;    // 2048: GRU input x
static constexpr int WS_GI     = 6144;    // 6144: w_ih @ x + b_ih
static constexpr int WS_GH     = 12288;   // 6144: w_hh @ h0 + b_hh
static constexpr int WS_HNEW   = 18432;   // 2048
static constexpr int WS_LOGITS = 20480;   // 50257
static constexpr int WS_LSE    = 70740;   // 1

// d_out layout: [0,VOC) log-probs | [VOC,VOC+HD) h_new | [VOC+HD,VOC+HD+SEQ) attn_weights

// ---------------------------------------------------------------------------
// Stage 1: embedding gather, attention scores + softmax, attention apply.
// Single workgroup; 1.5 MB of traffic, VALU is fine here.
// ---------------------------------------------------------------------------
__global__ void attn_kernel(const int* __restrict__ word,
                            const float* __restrict__ hidden,
                            const float* __restrict__ enc,
                            const float* __restrict__ emb,
                            const float* __restrict__ attn_w,
                            const float* __restrict__ attn_b,
                            float* __restrict__ ws,
                            float* __restrict__ out) {
    __shared__ float cat[2 * HD];   // [embedded | h0]
    __shared__ float red[256];
    __shared__ float sw[SEQ];
    const int tid = threadIdx.x;
    const int w = word[0];

    for (int i = tid; i < HD; i += 256) {
        float e = emb[(size_t)w * HD + i];
        cat[i] = e;
        cat[HD + i] = hidden[i];
        ws[WS_XCAT + i] = e;        // first half of concat for combine stage
    }
    __syncthreads();

    // scores[s] = attn_w[s,:] . cat + attn_b[s]; 4 threads per s over K=4096
    {
        const int s = tid & 63;
        const int chunk = tid >> 6;
        const float* wr = attn_w + (size_t)s * (2 * HD) + chunk * 1024;
        const float* cv = cat + chunk * 1024;
        float acc = 0.f;
        #pragma unroll 4
        for (int k = 0; k < 1024; ++k) acc += wr[k] * cv[k];
        red[tid] = acc;
    }
    __syncthreads();
    if (tid < SEQ)
        sw[tid] = red[tid] + red[64 + tid] + red[128 + tid] + red[192 + tid] + attn_b[tid];
    __syncthreads();

    if (tid == 0) {                  // 64-wide softmax, serial is trivial
        float m = sw[0];
        for (int i = 1; i < SEQ; ++i) m = fmaxf(m, sw[i]);
        float sum = 0.f;
        for (int i = 0; i < SEQ; ++i) { float e = expf(sw[i] - m); sw[i] = e; sum += e; }
        red[0] = 1.0f / sum;
    }
    __syncthreads();
    const float inv = red[0];
    if (tid < SEQ) {
        float wn = sw[tid] * inv;
        sw[tid] = wn;
        out[VOC + HD + tid] = wn;    // attn_weights output
    }
    __syncthreads();

    // attn_applied[h] = sum_s sw[s] * enc[s][h]  (coalesced over h)
    for (int h = tid; h < HD; h += 256) {
        float a = 0.f;
        #pragma unroll 8
        for (int s = 0; s < SEQ; ++s) a += sw[s] * enc[(size_t)s * HD + h];
        ws[WS_XCAT + HD + h] = a;    // second half of concat
    }
}

// ---------------------------------------------------------------------------
// WMMA-based GEMV: out[r] = W[r,:] . v + bias[r], optional ReLU.
// Each wave computes 16 rows with V_WMMA_F32_16X16X4_F32:
//   A (16x4) = 16 weight rows x 4 K values; B (4x16) = v broadcast to all cols.
// 4 waves / block => 64 rows per block. K must be a multiple of 8.
// ---------------------------------------------------------------------------
__global__ void gemv16_kernel(const float* __restrict__ W,
                              const float* __restrict__ v,
                              const float* __restrict__ bias,
                              float* __restrict__ out,
                              int rows, int K, int relu) {
    __shared__ float vlds[4096];
    const int tid = threadIdx.x;
    for (int i = tid; i < K; i += 128) vlds[i] = v[i];
    __syncthreads();

    const int lane = tid & 31;
    const int wv = tid >> 5;
    const int r0 = (blockIdx.x * 4 + wv) * 16;
    const int m = lane & 15;
    const int kh = lane >> 4;                // K-half select (A layout)
    int row = r0 + m;
    int rowc = row < rows ? row : rows - 1;  // clamp: keep EXEC all-ones
    const float* wrow = W + (size_t)rowc * K + 2 * kh;
    const float* vl = vlds + 2 * kh;

    v8f acc0 = {0.f, 0.f, 0.f, 0.f, 0.f, 0.f, 0.f, 0.f};
    v8f acc1 = {0.f, 0.f, 0.f, 0.f, 0.f, 0.f, 0.f, 0.f};
    #pragma unroll 2
    for (int kb = 0; kb < K; kb += 8) {
        v2f a0 = *(const v2f*)(wrow + kb);
        v2f b0 = *(const v2f*)(vl + kb);
        acc0 = __builtin_amdgcn_wmma_f32_16x16x4_f32(
            false, a0, false, b0, (short)0, acc0, false, false);
        v2f a1 = *(const v2f*)(wrow + kb + 4);
        v2f b1 = *(const v2f*)(vl + kb + 4);
        acc1 = __builtin_amdgcn_wmma_f32_16x16x4_f32(
            false, a1, false, b1, (short)0, acc1, false, false);
    }
    v8f acc = acc0 + acc1;

    // C/D layout: lane 0 holds N=0, M=0..7 in vgpr 0..7; lane 16 holds M=8..15
    if ((lane & 15) == 0) {
        const int base = r0 + kh * 8;
        #pragma unroll
        for (int g = 0; g < 8; ++g) {
            int r = base + g;
            if (r < rows) {
                float val = acc[g] + bias[r];
                if (relu) val = fmaxf(val, 0.f);
                out[r] = val;
            }
        }
    }
}

// ---------------------------------------------------------------------------
// GRU cell combine (gate order r, z, n)
// ---------------------------------------------------------------------------
__global__ void gru_kernel(const float* __restrict__ hidden,
                           const float* __restrict__ gi,
                           const float* __restrict__ gh,
                           float* __restrict__ hnew_ws,
                           float* __restrict__ out) {
    int i = blockIdx.x * blockDim.x + threadIdx.x;
    if (i >= HD) return;
    float i_r = gi[i], i_z = gi[HD + i], i_n = gi[2 * HD + i];
    float h_r = gh[i], h_z = gh[HD + i], h_n = gh[2 * HD + i];
    float r = 1.f / (1.f + expf(-(i_r + h_r)));
    float z = 1.f / (1.f + expf(-(i_z + h_z)));
    float n = tanhf(i_n + r * h_n);
    float h0 = hidden[i];
    float hn = (1.f - z) * n + z * h0;
    hnew_ws[i] = hn;
    out[VOC + i] = hn;               // h_new output
}

// ---------------------------------------------------------------------------
// log-sum-exp over the 50257 logits (single block, two passes; 400 KB total)
// ---------------------------------------------------------------------------
__global__ void lse_kernel(const float* __restrict__ logits,
                           float* __restrict__ lse) {
    __shared__ float red[1024];
    const int tid = threadIdx.x;
    float m = -INFINITY;
    for (int i = tid; i < VOC; i += 1024) m = fmaxf(m, logits[i]);
    red[tid] = m;
    __syncthreads();
    for (int s = 512; s > 0; s >>= 1) {
        if (tid < s) red[tid] = fmaxf(red[tid], red[tid + s]);
        __syncthreads();
    }
    m = red[0];
    __syncthreads();
    float sum = 0.f;
    for (int i = tid; i < VOC; i += 1024) sum += expf(logits[i] - m);
    red[tid] = sum;
    __syncthreads();
    for (int s = 512; s > 0; s >>= 1) {
        if (tid < s) red[tid] += red[tid + s];
        __syncthreads();
    }
    if (tid == 0) *lse = m + logf(red[0]);
}

__global__ void finalize_kernel(const float* __restrict__ logits,
                                const float* __restrict__ lse,
                                float* __restrict__ out) {
    int i = blockIdx.x * 256 + threadIdx.x;
    if (i < VOC) out[i] = logits[i] - *lse;
}

// ---------------------------------------------------------------------------
extern "C" void kernel_launch(void* const* d_in, const int* in_sizes, int n_in,
                              void* d_out, int out_size, void* d_ws, size_t ws_size,
                              hipStream_t stream) {
    const int*   word    = (const int*)  d_in[0];
    const float* hidden  = (const float*)d_in[1];
    const float* enc     = (const float*)d_in[2];
    const float* emb     = (const float*)d_in[3];
    const float* attn_w  = (const float*)d_in[4];
    const float* attn_b  = (const float*)d_in[5];
    const float* comb_w  = (const float*)d_in[6];
    const float* comb_b  = (const float*)d_in[7];
    const float* w_ih    = (const float*)d_in[8];
    const float* w_hh    = (const float*)d_in[9];
    const float* b_ih    = (const float*)d_in[10];
    const float* b_hh    = (const float*)d_in[11];
    const float* out_w   = (const float*)d_in[12];
    const float* out_b   = (const float*)d_in[13];
    float* out = (float*)d_out;
    float* ws  = (float*)d_ws;

    // 1) embedding + attention + attn_applied -> ws[XCAT..XCAT+4096)
    attn_kernel<<<1, 256, 0, stream>>>(word, hidden, enc, emb, attn_w, attn_b, ws, out);
    // 2) x = relu(comb_w @ xcat + comb_b)   (2048 rows, K=4096)
    gemv16_kernel<<<(HD + 63) / 64, 128, 0, stream>>>(
        comb_w, ws + WS_XCAT, comb_b, ws + WS_XVEC, HD, 2 * HD, 1);
    // 3) gi = w_ih @ x + b_ih               (6144 rows, K=2048)
    gemv16_kernel<<<(3 * HD + 63) / 64, 128, 0, stream>>>(
        w_ih, ws + WS_XVEC, b_ih, ws + WS_GI, 3 * HD, HD, 0);
    // 4) gh = w_hh @ h0 + b_hh              (6144 rows, K=2048)
    gemv16_kernel<<<(3 * HD + 63) / 64, 128, 0, stream>>>(
        w_hh, hidden, b_hh, ws + WS_GH, 3 * HD, HD, 0);
    // 5) GRU combine -> h_new
    gru_kernel<<<(HD + 255) / 256, 256, 0, stream>>>(
        hidden, ws + WS_GI, ws + WS_GH, ws + WS_HNEW, out);
    // 6) logits = out_w @ h_new + out_b     (50257 rows, K=2048) — 412 MB, the hot loop
    gemv16_kernel<<<(VOC + 63) / 64, 128, 0, stream>>>(
        out_w, ws + WS_HNEW, out_b, ws + WS_LOGITS, VOC, HD, 0);
    // 7) log_softmax
    lse_kernel<<<1, 1024, 0, stream>>>(ws + WS_LOGITS, ws + WS_LSE);
    finalize_kernel<<<(VOC + 255) / 256, 256, 0, stream>>>(ws + WS_LOGITS, ws + WS_LSE, out);
}